// GIN2_26792005993051
// MI455X (gfx1250) — compile-verified
//
#include <hip/hip_runtime.h>
#include <hip/hip_bf16.h>

#define GDIM 128

typedef __attribute__((ext_vector_type(2))) float v2f;
typedef __attribute__((ext_vector_type(8))) float v8f;

// ---------------------------------------------------------------------------
// t[i] = h[i]  (float4 vectorized copy, initializes the aggregation buffer)
// ---------------------------------------------------------------------------
__global__ __launch_bounds__(256) void copy_f4(const float4* __restrict__ in,
                                               float4* __restrict__ out, int n4) {
    int i = blockIdx.x * blockDim.x + threadIdx.x;
    if (i < n4) out[i] = in[i];
}

// ---------------------------------------------------------------------------
// t[dst[e]] += h[src[e]]   (one wave32 per edge, float4 per lane -> 128 feats)
// ---------------------------------------------------------------------------
__global__ __launch_bounds__(256) void scatter_add(const float* __restrict__ h,
                                                   const int* __restrict__ src,
                                                   const int* __restrict__ dst,
                                                   float* __restrict__ t, int nE) {
    int tid = blockIdx.x * blockDim.x + threadIdx.x;
    int edge = tid >> 5;
    int lane = tid & 31;
    if (edge >= nE) return;
    int s = src[edge];
    int d = dst[edge];
    const float4 v = *(const float4*)(h + (size_t)s * GDIM + lane * 4);
    float* tp = t + (size_t)d * GDIM + lane * 4;
    atomicAdd(tp + 0, v.x);
    atomicAdd(tp + 1, v.y);
    atomicAdd(tp + 2, v.z);
    atomicAdd(tp + 3, v.w);
}

// ---------------------------------------------------------------------------
// out = relu( bn?( X @ W + bias ) )   X:[nrows x 128]  W:[128 x 128] row-major
//
// Block = 256 threads = 8 waves arranged 4 (M) x 2 (N); block tile 128x128.
// Each wave computes 32 rows x 64 cols = 2 M-tiles x 4 N-tiles (8 v8f accs).
// W staged in LDS in k-PAIRED layout: sW2[k/2][n] = {W[k][n], W[k+1][n]},
// so each B fragment is one aligned ds_load_b64 (no VGPR glue moves).
// fp32 WMMA 16x16x4: A 16x4 (2 VGPR), B 4x16 (2 VGPR), C/D 16x16 (8 VGPR).
// ---------------------------------------------------------------------------
__global__ __launch_bounds__(256) void gin_gemm(const float* __restrict__ X,
                                                const float* __restrict__ W,
                                                const float* __restrict__ bias,
                                                const float* __restrict__ g,
                                                const float* __restrict__ beta,
                                                const float* __restrict__ rm,
                                                const float* __restrict__ rv,
                                                int use_bn,
                                                float* __restrict__ out, int nrows) {
    __shared__ v2f sW2[(GDIM / 2) * GDIM];  // 64 KB, k-paired weight tile
    {
        // Coalesced float4 reads of row-major W, scattered b32 writes into
        // the paired layout (one-time 64 KB stage).
        const float4* Wv = (const float4*)W;
        float* sWf = (float*)sW2;
        for (int idx = threadIdx.x; idx < GDIM * (GDIM / 4); idx += 256) {
            int k = idx >> 5;           // row of W
            int n0 = (idx & 31) * 4;    // first of 4 columns
            float4 v = Wv[idx];
            int kp = k >> 1, comp = k & 1;
            const float* vp = (const float*)&v;
#pragma unroll
            for (int j = 0; j < 4; ++j)
                sWf[((kp * GDIM) + n0 + j) * 2 + comp] = vp[j];
        }
    }
    __syncthreads();

    const int lane = threadIdx.x & 31;
    const int wave = threadIdx.x >> 5;
    const int wrow = wave >> 1;   // 0..3
    const int wcol = wave & 1;    // 0..1
    const int rowBase = blockIdx.x * 128 + wrow * 32;
    const int colBase = wcol * 64;
    const int mrow = lane & 15;       // M index within A fragment
    const int khalf = lane >> 4;      // lanes 0-15: K pair 0 ; 16-31: K pair 1
    const int ncol = lane & 15;       // N index within B/C/D fragments

    int arow0 = rowBase + mrow;
    int arow1 = rowBase + 16 + mrow;
    if (arow0 >= nrows) arow0 = nrows - 1;  // clamp loads; stores masked below
    if (arow1 >= nrows) arow1 = nrows - 1;
    const float* Ar0 = X + (size_t)arow0 * GDIM;
    const float* Ar1 = X + (size_t)arow1 * GDIM;
    __builtin_prefetch(Ar0, 0, 0);   // global_prefetch_b8
    __builtin_prefetch(Ar1, 0, 0);

    v8f acc[2][4];
#pragma unroll
    for (int mi = 0; mi < 2; ++mi)
#pragma unroll
        for (int nt = 0; nt < 4; ++nt)
            acc[mi][nt] = (v8f){0.f, 0.f, 0.f, 0.f, 0.f, 0.f, 0.f, 0.f};

#pragma unroll 4
    for (int k0 = 0; k0 < GDIM; k0 += 4) {
        const int koff = khalf * 2;
        v2f a0, a1;
        a0.x = Ar0[k0 + koff + 0];
        a0.y = Ar0[k0 + koff + 1];
        a1.x = Ar1[k0 + koff + 0];
        a1.y = Ar1[k0 + koff + 1];
        const int kp = (k0 >> 1) + khalf;
        v2f b[4];
#pragma unroll
        for (int nt = 0; nt < 4; ++nt)
            b[nt] = sW2[kp * GDIM + colBase + nt * 16 + ncol];
#pragma unroll
        for (int nt = 0; nt < 4; ++nt) {
            acc[0][nt] = __builtin_amdgcn_wmma_f32_16x16x4_f32(
                false, a0, false, b[nt], (short)0, acc[0][nt], false, false);
            acc[1][nt] = __builtin_amdgcn_wmma_f32_16x16x4_f32(
                false, a1, false, b[nt], (short)0, acc[1][nt], false, false);
        }
    }

    // Epilogue: C/D layout -> lane<16: M=r, lane>=16: M=r+8 ; N = lane&15
    const int mAdd = (lane < 16) ? 0 : 8;
#pragma unroll
    for (int nt = 0; nt < 4; ++nt) {
        const int n = colBase + nt * 16 + ncol;
        float bi = bias[n];
        float scale = 1.0f, shift = 0.0f;
        if (use_bn) {
            float inv = rsqrtf(rv[n] + 1e-5f);
            scale = g[n] * inv;
            shift = beta[n] - rm[n] * scale;
        }
#pragma unroll
        for (int mi = 0; mi < 2; ++mi) {
#pragma unroll
            for (int r = 0; r < 8; ++r) {
                int m = rowBase + mi * 16 + r + mAdd;
                if (m < nrows) {
                    float v = acc[mi][nt][r] + bi;
                    if (use_bn) v = v * scale + shift;
                    v = fmaxf(v, 0.0f);
                    out[(size_t)m * GDIM + n] = v;
                }
            }
        }
    }
}

// ---------------------------------------------------------------------------
// e[edge] = concat(h[src], h[dst]);  out[edge] = e @ cls_W + cls_b  (NC=6)
// wave32 per edge; per-lane FMAs + shuffle reduction (memory-bound op).
// ---------------------------------------------------------------------------
__global__ __launch_bounds__(256) void edge_cls(const float* __restrict__ h,
                                                const int* __restrict__ src,
                                                const int* __restrict__ dst,
                                                const float* __restrict__ Wc,
                                                const float* __restrict__ bc,
                                                float* __restrict__ outv,
                                                float* __restrict__ e, int nE) {
    int tid = blockIdx.x * blockDim.x + threadIdx.x;
    int edge = tid >> 5;
    int lane = tid & 31;
    if (edge >= nE) return;
    int s = src[edge];
    int d = dst[edge];
    const float4 hs = *(const float4*)(h + (size_t)s * GDIM + lane * 4);
    const float4 hd = *(const float4*)(h + (size_t)d * GDIM + lane * 4);
    float4* ev = (float4*)(e + (size_t)edge * (2 * GDIM));
    ev[lane] = hs;
    ev[32 + lane] = hd;

    float p[6];
#pragma unroll
    for (int c = 0; c < 6; ++c) p[c] = 0.0f;
    const float* hsp = (const float*)&hs;
    const float* hdp = (const float*)&hd;
#pragma unroll
    for (int j = 0; j < 4; ++j) {
        int f0 = lane * 4 + j;
        float a = hsp[j];
        float b = hdp[j];
#pragma unroll
        for (int c = 0; c < 6; ++c)
            p[c] += a * Wc[f0 * 6 + c] + b * Wc[(GDIM + f0) * 6 + c];
    }
#pragma unroll
    for (int c = 0; c < 6; ++c) {
        float v = p[c];
        for (int off = 16; off > 0; off >>= 1) v += __shfl_xor(v, off, 32);
        if (lane == 0) outv[(size_t)edge * 6 + c] = v + bc[c];
    }
}

// ---------------------------------------------------------------------------
extern "C" void kernel_launch(void* const* d_in, const int* in_sizes, int n_in,
                              void* d_out, int out_size, void* d_ws, size_t ws_size,
                              hipStream_t stream) {
    const float* x = (const float*)d_in[0];
    const int* eidx = (const int*)d_in[1];
    const int N = in_sizes[0] / GDIM;
    const int E = in_sizes[1] / 2;
    const int* src = eidx;
    const int* dst = eidx + E;

    const float* c1_W1 = (const float*)d_in[3];
    const float* c1_b1 = (const float*)d_in[4];
    const float* c1_g  = (const float*)d_in[5];
    const float* c1_be = (const float*)d_in[6];
    const float* c1_rm = (const float*)d_in[7];
    const float* c1_rv = (const float*)d_in[8];
    const float* c1_W2 = (const float*)d_in[9];
    const float* c1_b2 = (const float*)d_in[10];
    const float* cv_W1 = (const float*)d_in[11];
    const float* cv_b1 = (const float*)d_in[12];
    const float* cv_g  = (const float*)d_in[13];
    const float* cv_be = (const float*)d_in[14];
    const float* cv_rm = (const float*)d_in[15];
    const float* cv_rv = (const float*)d_in[16];
    const float* cv_W2 = (const float*)d_in[17];
    const float* cv_b2 = (const float*)d_in[18];
    const float* ln_W  = (const float*)d_in[19];
    const float* ln_b  = (const float*)d_in[20];
    const float* cl_W  = (const float*)d_in[21];
    const float* cl_b  = (const float*)d_in[22];

    float* bufT = (float*)d_ws;                    // aggregation target
    float* bufZ = bufT + (size_t)N * GDIM;         // MLP hidden
    float* bufH = bufZ + (size_t)N * GDIM;         // layer output

    dim3 blk(256);
    const int n4 = N * (GDIM / 4);
    const int copyBlocks = (n4 + 255) / 256;
    const int edgeBlocks = (int)(((long long)E * 32 + 255) / 256);
    const int gemmBlocks = (N + 127) / 128;

    // ---- conv1: GIN(x) ----
    copy_f4<<<copyBlocks, blk, 0, stream>>>((const float4*)x, (float4*)bufT, n4);
    scatter_add<<<edgeBlocks, blk, 0, stream>>>(x, src, dst, bufT, E);
    gin_gemm<<<gemmBlocks, blk, 0, stream>>>(bufT, c1_W1, c1_b1, c1_g, c1_be, c1_rm, c1_rv, 1, bufZ, N);
    gin_gemm<<<gemmBlocks, blk, 0, stream>>>(bufZ, c1_W2, c1_b2, nullptr, nullptr, nullptr, nullptr, 0, bufH, N);

    // ---- conv2, conv3 ----
    for (int i = 0; i < 2; ++i) {
        const size_t wo = (size_t)i * GDIM * GDIM;
        const size_t vo = (size_t)i * GDIM;
        copy_f4<<<copyBlocks, blk, 0, stream>>>((const float4*)bufH, (float4*)bufT, n4);
        scatter_add<<<edgeBlocks, blk, 0, stream>>>(bufH, src, dst, bufT, E);
        gin_gemm<<<gemmBlocks, blk, 0, stream>>>(bufT, cv_W1 + wo, cv_b1 + vo, cv_g + vo, cv_be + vo,
                                                 cv_rm + vo, cv_rv + vo, 1, bufZ, N);
        gin_gemm<<<gemmBlocks, blk, 0, stream>>>(bufZ, cv_W2 + wo, cv_b2 + vo, nullptr, nullptr, nullptr,
                                                 nullptr, 0, bufH, N);
    }

    // ---- lin1, lin2 ----
    gin_gemm<<<gemmBlocks, blk, 0, stream>>>(bufH, ln_W, ln_b, nullptr, nullptr, nullptr, nullptr, 0, bufZ, N);
    gin_gemm<<<gemmBlocks, blk, 0, stream>>>(bufZ, ln_W + (size_t)GDIM * GDIM, ln_b + GDIM, nullptr, nullptr,
                                             nullptr, nullptr, 0, bufH, N);

    // ---- classifier over edges; d_out = [out (E*6) | e (E*256)] ----
    float* out0 = (float*)d_out;
    float* eout = out0 + (size_t)E * 6;
    edge_cls<<<edgeBlocks, blk, 0, stream>>>(bufH, src, dst, cl_W, cl_b, out0, eout, E);
}